// PhasePromptGenerator_87351044866819
// MI455X (gfx1250) — compile-verified
//
#include <hip/hip_runtime.h>
#include <math.h>

// ---------------------------------------------------------------------------
// PhasePromptGenerator for MI455X (gfx1250, wave32).
// FFT expressed as dense f32 GEMMs with the symmetric DFT matrix, executed
// with V_WMMA_F32_16X16X4_F32.  All intermediates live in d_ws (~70 MB, fits
// in the 192 MB L2).
// ---------------------------------------------------------------------------

typedef __attribute__((ext_vector_type(2))) float v2f;
typedef __attribute__((ext_vector_type(8))) float v8f;

#define NN 512
#define BATCH 16
#define IMG_PIX (NN * NN)           // 262144
#define TOTAL_PIX (BATCH * IMG_PIX) // 4194304
#define HIST_BINS 4096
#define TOPK 5
#define KTH 26214                   // int((1-0.9)*512*512)

// ---------------------------------------------------------------------------
// Twiddle + Hann init.  e^{-2*pi*i*k*n/N} depends only on (k*n) mod N, so the
// angle is computed exactly in f32.
// ---------------------------------------------------------------------------
__global__ void tw_init_kernel(float* __restrict__ Wr, float* __restrict__ Wi,
                               float* __restrict__ hann) {
    int idx = blockIdx.x * blockDim.x + threadIdx.x;
    if (idx < NN) {
        hann[idx] = 0.5f * (1.0f - cosf(6.2831853071795864f * (float)idx / 511.0f));
    }
    if (idx < NN * NN) {
        int k = idx >> 9;
        int n = idx & 511;
        int m = (k * n) & 511;
        float ang = -6.2831853071795864f * (float)m / 512.0f;
        Wr[idx] = cosf(ang);
        Wi[idx] = sinf(ang);
    }
}

// ---------------------------------------------------------------------------
// Complex GEMM:  C = A * B  (512x512, row-major), batch via grid.z.
//   Cr = Ar*Br - (aiSign*Ai)*(biSign*Bi)
//   Ci = Ar*(biSign*Bi) + (aiSign*Ai)*Br
// Ai / Bi may be nullptr (treated as zero).  applyWin multiplies A elements by
// hann[row]*hann[col] on load (stage-1 window fusion).
//
// Block: 256 threads = 8 waves, C tile 32(M) x 64(N), waves in a 2x4 grid of
// 16x16 sub-tiles.  K staged through LDS in chunks of 32; each wave issues
// 4 v_wmma_f32_16x16x4_f32 per K=4 step.
// ---------------------------------------------------------------------------
__global__ __launch_bounds__(256)
void cgemm_kernel(const float* __restrict__ Ar, const float* __restrict__ Ai,
                  const float* __restrict__ Br, const float* __restrict__ Bi,
                  float* __restrict__ Cr, float* __restrict__ Ci,
                  int batchA, int batchB, float aiSign, float biSign,
                  int applyWin, const float* __restrict__ hann) {
    __shared__ float sAr[32][33];
    __shared__ float sAi[32][33];
    __shared__ float sBr[32][64];
    __shared__ float sBi[32][64];

    const int tid = threadIdx.x;
    const int m0  = blockIdx.y * 32;
    const int n0  = blockIdx.x * 64;
    const int z   = blockIdx.z;
    const size_t aOff = batchA ? (size_t)z * IMG_PIX : 0;
    const size_t bOff = batchB ? (size_t)z * IMG_PIX : 0;
    const size_t cOff = (size_t)z * IMG_PIX;

    const int wave = tid >> 5;
    const int lane = tid & 31;
    const int wM = wave >> 2;            // 0..1
    const int wN = wave & 3;             // 0..3
    const int aRow = wM * 16 + (lane & 15);
    const int bCol = wN * 16 + (lane & 15);
    const int kh   = (lane >> 4) << 1;   // 0 or 2 (K half select, ISA 7.12.2)

    v8f accR = {};
    v8f accI = {};

    for (int kc = 0; kc < NN; kc += 32) {
        // ---- stage A chunk (32x32) into LDS, optional window fusion ----
#pragma unroll
        for (int i = 0; i < 4; ++i) {
            int e = tid + i * 256;
            int r = e >> 5, c = e & 31;
            int gr = m0 + r, gc = kc + c;
            float vr = Ar[aOff + (size_t)gr * NN + gc];
            float vi = Ai ? Ai[aOff + (size_t)gr * NN + gc] * aiSign : 0.0f;
            if (applyWin) {
                float w = hann[gr] * hann[gc];
                vr *= w; vi *= w;
            }
            sAr[r][c] = vr;
            sAi[r][c] = vi;
        }
        // ---- stage B chunk (32x64) into LDS ----
#pragma unroll
        for (int i = 0; i < 8; ++i) {
            int e = tid + i * 256;
            int r = e >> 6, c = e & 63;
            int gr = kc + r, gc = n0 + c;
            sBr[r][c] = Br[bOff + (size_t)gr * NN + gc];
            sBi[r][c] = Bi ? Bi[bOff + (size_t)gr * NN + gc] * biSign : 0.0f;
        }
        if (kc + 32 < NN) {  // hint next chunk into cache (global_prefetch_b8)
            __builtin_prefetch(&Br[bOff + (size_t)(kc + 32) * NN + n0], 0, 1);
            __builtin_prefetch(&Ar[aOff + (size_t)(m0)*NN + (kc + 32)], 0, 1);
        }
        __syncthreads();

        // ---- 8 x (K=4) WMMA steps; complex mult = 4 real WMMAs ----
#pragma unroll
        for (int kk = 0; kk < 32; kk += 4) {
            v2f ar, ai2, ain, br, bi2;
            ar.x  = sAr[aRow][kk + kh];     ar.y  = sAr[aRow][kk + kh + 1];
            ai2.x = sAi[aRow][kk + kh];     ai2.y = sAi[aRow][kk + kh + 1];
            br.x  = sBr[kk + kh][bCol];     br.y  = sBr[kk + kh + 1][bCol];
            bi2.x = sBi[kk + kh][bCol];     bi2.y = sBi[kk + kh + 1][bCol];
            ain.x = -ai2.x;                 ain.y = -ai2.y;
            accR = __builtin_amdgcn_wmma_f32_16x16x4_f32(false, ar,  false, br,
                                                         (short)0, accR, false, false);
            accR = __builtin_amdgcn_wmma_f32_16x16x4_f32(false, ain, false, bi2,
                                                         (short)0, accR, false, false);
            accI = __builtin_amdgcn_wmma_f32_16x16x4_f32(false, ar,  false, bi2,
                                                         (short)0, accI, false, false);
            accI = __builtin_amdgcn_wmma_f32_16x16x4_f32(false, ai2, false, br,
                                                         (short)0, accI, false, false);
        }
        __syncthreads();
    }

    // ---- store C tile (f32 C/D layout: VGPR r -> M=r (lanes 0-15), M=r+8) ----
    const int colC  = n0 + wN * 16 + (lane & 15);
    const int rBase = m0 + wM * 16 + ((lane >> 4) << 3);
#pragma unroll
    for (int r = 0; r < 8; ++r) {
        size_t idx = cOff + (size_t)(rBase + r) * NN + colC;
        Cr[idx] = accR[r];
        Ci[idx] = accI[r];
    }
}

// ---------------------------------------------------------------------------
// Phase-only spectrum: z / |z| (angle(0) -> 1+0i, matching exp(i*angle(0))).
// ---------------------------------------------------------------------------
__global__ void phase_kernel(float* __restrict__ re, float* __restrict__ im, int n) {
    int idx = blockIdx.x * blockDim.x + threadIdx.x;
    if (idx >= n) return;
    float r = re[idx], i = im[idx];
    float m2 = r * r + i * i;
    if (m2 > 0.0f) {
        float inv = rsqrtf(m2);
        re[idx] = r * inv;
        im[idx] = i * inv;
    } else {
        re[idx] = 1.0f;
        im[idx] = 0.0f;
    }
}

// |recon|^2 (global positive scale is irrelevant: min-max norm cancels it).
__global__ void mag_kernel(const float* __restrict__ re, const float* __restrict__ im,
                           float* __restrict__ out, int n) {
    int idx = blockIdx.x * blockDim.x + threadIdx.x;
    if (idx >= n) return;
    float r = re[idx], i = im[idx];
    out[idx] = r * r + i * i;
}

// 5x5 Gaussian blur, sigma=1, zero padding.
__global__ void blur_kernel(const float* __restrict__ in, float* __restrict__ out) {
    int idx = blockIdx.x * blockDim.x + threadIdx.x;
    if (idx >= TOTAL_PIX) return;
    int b = idx >> 18;
    int rem = idx & (IMG_PIX - 1);
    int y = rem >> 9, x = rem & 511;
    const float w[5] = {0.05448868f, 0.24420135f, 0.40261995f, 0.24420135f, 0.05448868f};
    const float* p = in + (size_t)b * IMG_PIX;
    float acc = 0.0f;
#pragma unroll
    for (int dy = -2; dy <= 2; ++dy) {
        int yy = y + dy;
        if (yy < 0 || yy >= NN) continue;
#pragma unroll
        for (int dx = -2; dx <= 2; ++dx) {
            int xx = x + dx;
            if (xx < 0 || xx >= NN) continue;
            acc += w[dy + 2] * w[dx + 2] * p[yy * NN + xx];
        }
    }
    out[idx] = acc;
}

// Per-image min/max (one block per image).
__global__ __launch_bounds__(256)
void minmax_kernel(const float* __restrict__ in, float* __restrict__ mn,
                   float* __restrict__ mx) {
    __shared__ float smn[256], smx[256];
    int b = blockIdx.x, tid = threadIdx.x;
    const float* p = in + (size_t)b * IMG_PIX;
    float lmn = 3.4e38f, lmx = -3.4e38f;
    for (int i = tid; i < IMG_PIX; i += 256) {
        float v = p[i];
        lmn = fminf(lmn, v);
        lmx = fmaxf(lmx, v);
    }
    smn[tid] = lmn; smx[tid] = lmx;
    __syncthreads();
    for (int s = 128; s > 0; s >>= 1) {
        if (tid < s) {
            smn[tid] = fminf(smn[tid], smn[tid + s]);
            smx[tid] = fmaxf(smx[tid], smx[tid + s]);
        }
        __syncthreads();
    }
    if (tid == 0) { mn[b] = smn[0]; mx[b] = smx[0]; }
}

__global__ void zero_hist_kernel(int* __restrict__ hist) {
    int idx = blockIdx.x * blockDim.x + threadIdx.x;
    if (idx < BATCH * HIST_BINS) hist[idx] = 0;
}

// Normalize + border mask, write final sal, build per-image histogram.
__global__ void norm_kernel(const float* __restrict__ in, const float* __restrict__ mn,
                            const float* __restrict__ mx, float* __restrict__ sal,
                            int* __restrict__ hist) {
    int idx = blockIdx.x * blockDim.x + threadIdx.x;
    if (idx >= TOTAL_PIX) return;
    int b = idx >> 18;
    int rem = idx & (IMG_PIX - 1);
    int y = rem >> 9, x = rem & 511;
    float v = (in[idx] - mn[b]) / (mx[b] - mn[b] + 1e-8f);
    bool inB = (y >= 12) && (y < NN - 12) && (x >= 12) && (x < NN - 12);
    v = inB ? v : 0.0f;
    sal[idx] = v;
    int bin = (int)(v * (float)HIST_BINS);
    bin = bin < 0 ? 0 : (bin > HIST_BINS - 1 ? HIST_BINS - 1 : bin);
    atomicAdd(&hist[b * HIST_BINS + bin], 1);
}

// Histogram quantile: threshold = lower edge of bin holding the KTH-largest,
// then max(thr, 0.1).
__global__ void thr_kernel(const int* __restrict__ hist, float* __restrict__ thr) {
    int b = threadIdx.x;
    if (b >= BATCH) return;
    int cum = 0;
    float t = 0.0f;
    for (int bin = HIST_BINS - 1; bin >= 0; --bin) {
        cum += hist[b * HIST_BINS + bin];
        if (cum >= KTH) { t = (float)bin / (float)HIST_BINS; break; }
    }
    thr[b] = fmaxf(t, 0.1f);
}

// Greedy min-dist peak selection: 5 rounds of masked global argmax
// (equivalent to the reference's top-80 + greedy keep).
__global__ __launch_bounds__(256)
void select_kernel(const float* __restrict__ sal, const float* __restrict__ thr,
                   float* __restrict__ coords, float* __restrict__ labels) {
    __shared__ float sv[256];
    __shared__ int   si[256];
    __shared__ float kx[TOPK], ky[TOPK];
    __shared__ int   cnt;
    int b = blockIdx.x, tid = threadIdx.x;
    const float* p = sal + (size_t)b * IMG_PIX;
    float t = thr[b];
    if (tid == 0) cnt = 0;
    __syncthreads();

    for (int it = 0; it < TOPK; ++it) {
        int c = cnt;
        float best = -1.0f;   // valid scores >= thr >= 0.1
        int bidx = -1;
        for (int i = tid; i < IMG_PIX; i += 256) {
            float v = p[i];
            if (v < t) continue;
            int y = i >> 9, x = i & 511;
            bool near = false;
            for (int j = 0; j < c; ++j) {
                float dx = (float)x - kx[j];
                float dy = (float)y - ky[j];
                if (dx * dx + dy * dy < 100.0f) { near = true; break; }
            }
            if (near) continue;
            if (v > best || (v == best && (bidx < 0 || i < bidx))) { best = v; bidx = i; }
        }
        sv[tid] = best; si[tid] = bidx;
        __syncthreads();
        for (int s = 128; s > 0; s >>= 1) {
            if (tid < s) {
                bool take = (sv[tid + s] > sv[tid]) ||
                            (sv[tid + s] == sv[tid] && si[tid + s] >= 0 &&
                             (si[tid] < 0 || si[tid + s] < si[tid]));
                if (take) { sv[tid] = sv[tid + s]; si[tid] = si[tid + s]; }
            }
            __syncthreads();
        }
        if (tid == 0 && si[0] >= 0) {
            kx[cnt] = (float)(si[0] & 511);
            ky[cnt] = (float)(si[0] >> 9);
            cnt = cnt + 1;
        }
        __syncthreads();
    }

    if (tid == 0) {
        for (int i = 0; i < TOPK; ++i) {
            bool ok = i < cnt;
            coords[b * (TOPK * 2) + i * 2 + 0] = ok ? kx[i] : -1.0f;
            coords[b * (TOPK * 2) + i * 2 + 1] = ok ? ky[i] : -1.0f;
            labels[b * TOPK + i] = ok ? 1.0f : -1.0f;
        }
    }
}

// ---------------------------------------------------------------------------
// Launch: window+FFT (2 cgemm) -> phase -> iFFT (2 cgemm) -> |.|^2 -> blur ->
// minmax -> normalize+hist -> threshold -> select.
// ---------------------------------------------------------------------------
extern "C" void kernel_launch(void* const* d_in, const int* in_sizes, int n_in,
                              void* d_out, int out_size, void* d_ws, size_t ws_size,
                              hipStream_t stream) {
    (void)in_sizes; (void)n_in; (void)out_size; (void)ws_size;
    const float* img = (const float*)d_in[0];
    float* out = (float*)d_out;

    char* ws = (char*)d_ws;
    size_t off = 0;
    auto take = [&](size_t bytes) -> char* {
        char* p = ws + off;
        off = (off + bytes + 255) & ~(size_t)255;
        return p;
    };
    float* hann = (float*)take(NN * sizeof(float));
    float* Wr   = (float*)take((size_t)IMG_PIX * sizeof(float));
    float* Wi   = (float*)take((size_t)IMG_PIX * sizeof(float));
    float* b1r  = (float*)take((size_t)TOTAL_PIX * sizeof(float));
    float* b1i  = (float*)take((size_t)TOTAL_PIX * sizeof(float));
    float* b2r  = (float*)take((size_t)TOTAL_PIX * sizeof(float));
    float* b2i  = (float*)take((size_t)TOTAL_PIX * sizeof(float));
    float* mn   = (float*)take(BATCH * sizeof(float));
    float* mx   = (float*)take(BATCH * sizeof(float));
    float* thrb = (float*)take(BATCH * sizeof(float));
    int*   hist = (int*)take((size_t)BATCH * HIST_BINS * sizeof(int));

    float* coords = out;                       // 16*5*2
    float* labels = out + BATCH * TOPK * 2;    // 16*5
    float* sal    = out + BATCH * TOPK * 3;    // 16*1*512*512

    tw_init_kernel<<<(IMG_PIX + 255) / 256, 256, 0, stream>>>(Wr, Wi, hann);

    dim3 gg(NN / 64, NN / 32, BATCH);
    // F1 = (win .* X) * W          (A real, batched; window fused)
    cgemm_kernel<<<gg, 256, 0, stream>>>(img, nullptr, Wr, Wi, b1r, b1i,
                                         1, 0, 1.0f, 1.0f, 1, hann);
    // F = W * F1
    cgemm_kernel<<<gg, 256, 0, stream>>>(Wr, Wi, b1r, b1i, b2r, b2i,
                                         0, 1, 1.0f, 1.0f, 0, hann);
    // P = F / |F|
    phase_kernel<<<TOTAL_PIX / 256, 256, 0, stream>>>(b2r, b2i, TOTAL_PIX);
    // T = conj(W) * P
    cgemm_kernel<<<gg, 256, 0, stream>>>(Wr, Wi, b2r, b2i, b1r, b1i,
                                         0, 1, -1.0f, 1.0f, 0, hann);
    // R = T * conj(W)      (recon up to a positive scale)
    cgemm_kernel<<<gg, 256, 0, stream>>>(b1r, b1i, Wr, Wi, b2r, b2i,
                                         1, 0, 1.0f, -1.0f, 0, hann);
    // sal0 = |R|^2
    mag_kernel<<<TOTAL_PIX / 256, 256, 0, stream>>>(b2r, b2i, b1r, TOTAL_PIX);
    // blur
    blur_kernel<<<TOTAL_PIX / 256, 256, 0, stream>>>(b1r, b2r);
    // per-image min/max
    minmax_kernel<<<BATCH, 256, 0, stream>>>(b2r, mn, mx);
    // threshold histogram
    zero_hist_kernel<<<(BATCH * HIST_BINS) / 256, 256, 0, stream>>>(hist);
    norm_kernel<<<TOTAL_PIX / 256, 256, 0, stream>>>(b2r, mn, mx, sal, hist);
    thr_kernel<<<1, 32, 0, stream>>>(hist, thrb);
    // greedy NMS point selection
    select_kernel<<<BATCH, 256, 0, stream>>>(sal, thrb, coords, labels);
}